// GraphMessagePassing_24713241821695
// MI455X (gfx1250) — compile-verified
//
#include <hip/hip_runtime.h>

typedef __attribute__((ext_vector_type(2))) float v2f;
typedef __attribute__((ext_vector_type(8))) float v8f;

namespace {
constexpr int kNodes   = 50000;
constexpr int kEdges   = 800000;
constexpr int kNodeDim = 64;
constexpr int kEdgeDim = 32;
constexpr int kHidden  = 64;
constexpr int kLayers  = 2;
constexpr int kInMsg   = kNodeDim + kEdgeDim;  // 96
constexpr int kInUpd   = kNodeDim + kHidden;   // 128

constexpr int kWaves   = 4;            // waves per block
constexpr int kThreads = kWaves * 32;  // 128

// Padded LDS row strides (dwords) so 16 consecutive rows land in distinct
// banks of the 64-bank LDS.
constexpr int kW1Stride = kInMsg + 4;   // 100
constexpr int kW2Stride = kHidden + 4;  // 68
constexpr int kU1Stride = kInUpd + 4;   // 132
constexpr int kHStride  = kHidden + 4;  // 68
}  // namespace

// D = A(16x4, f32) * B(4x16, f32) + C(16x16, f32), exact fp32 on matrix pipe.
__device__ __forceinline__ v8f wmma_f32(v2f a, v2f b, v8f c) {
  return __builtin_amdgcn_wmma_f32_16x16x4_f32(false, a, false, b, (short)0, c,
                                               false, false);
}

__global__ void gmp_zero_kernel(float* __restrict__ p, int n) {
  int i = blockIdx.x * blockDim.x + threadIdx.x;
  if (i < n) p[i] = 0.0f;
}

__global__ void gmp_norm_kernel(float* __restrict__ agg,
                                const float* __restrict__ deg) {
  int i = blockIdx.x * blockDim.x + threadIdx.x;
  if (i < kNodes * kHidden) {
    float d = deg[i >> 6];
    agg[i] = agg[i] / (d + 1e-8f);
  }
}

// One wave handles 16 edges: msg = sum_l( relu(X@W1_l + b1_l) @ W2_l + b2_l ),
// then scatter-adds into agg[dst] and bumps deg[dst].
__global__ __launch_bounds__(kThreads) void gmp_edge_kernel(
    const float* __restrict__ node_f, const float* __restrict__ edge_f,
    const float* __restrict__ msg_w1, const float* __restrict__ msg_b1,
    const float* __restrict__ msg_w2, const float* __restrict__ msg_b2,
    const int* __restrict__ src, const int* __restrict__ dst,
    float* __restrict__ agg, float* __restrict__ deg) {
  __shared__ float sW1[kLayers][kHidden][kW1Stride];  // transposed: [l][n][k]
  __shared__ float sW2[kLayers][kHidden][kW2Stride];  // transposed: [l][n][k]
  __shared__ float sB1[kLayers][kHidden];
  __shared__ float sB2[kHidden];                      // b2[0]+b2[1]
  __shared__ float sH[kWaves][16][kHStride];          // relu(h) tile per wave
  __shared__ int   sDst[kWaves][16];

  const int tid = threadIdx.x;

  // Cooperative load + transpose of the message MLP weights into LDS.
  for (int i = tid; i < kLayers * kInMsg * kHidden; i += kThreads) {
    int l = i / (kInMsg * kHidden);
    int r = i - l * (kInMsg * kHidden);
    int k = r / kHidden;
    int n = r - k * kHidden;
    sW1[l][n][k] = msg_w1[i];  // global layout [l][k][n] flat == i
  }
  for (int i = tid; i < kLayers * kHidden * kHidden; i += kThreads) {
    int l = i / (kHidden * kHidden);
    int r = i - l * (kHidden * kHidden);
    int k = r / kHidden;
    int n = r - k * kHidden;
    sW2[l][n][k] = msg_w2[i];
  }
  if (tid < kLayers * kHidden) sB1[tid / kHidden][tid % kHidden] = msg_b1[tid];
  if (tid < kHidden) sB2[tid] = msg_b2[tid] + msg_b2[kHidden + tid];
  __syncthreads();

  const int wave  = tid >> 5;
  const int lane  = tid & 31;
  const int lrow  = lane & 15;   // M row (A/B) / N column owned by this lane
  const int khalf = lane >> 4;   // which K pair of the 16x16x4 fragment
  const int koff  = khalf * 2;

  const long ebase = ((long)blockIdx.x * kWaves + wave) * 16;

  const int sidx = src[ebase + lrow];
  if (lane < 16) {
    int d = dst[ebase + lane];
    sDst[wave][lane] = d;
    atomicAdd(&deg[d], 1.0f);
  }

  const float* nrow = node_f + (long)sidx * kNodeDim;          // gathered row
  const float* erow = edge_f + (ebase + lrow) * kEdgeDim;      // streamed row

  // Message accumulator across both layers, pre-seeded with b2[0]+b2[1].
  v8f macc[4];
#pragma unroll
  for (int nt = 0; nt < 4; ++nt) {
    float b = sB2[nt * 16 + lrow];
#pragma unroll
    for (int j = 0; j < 8; ++j) macc[nt][j] = b;
  }

#pragma unroll
  for (int l = 0; l < kLayers; ++l) {
    v8f hacc[4];
#pragma unroll
    for (int nt = 0; nt < 4; ++nt) {
      float b = sB1[l][nt * 16 + lrow];
#pragma unroll
      for (int j = 0; j < 8; ++j) hacc[nt][j] = b;
    }

    // GEMM1: [16 x 96] @ [96 x 64]; A straight from global (L2-hot gathers).
#pragma unroll
    for (int ks = 0; ks < kInMsg / 4; ++ks) {
      const int col = ks * 4 + koff;  // branch below is wave-uniform per ks
      v2f a;
      if (col < kNodeDim) {
        a.x = nrow[col];
        a.y = nrow[col + 1];
      } else {
        a.x = erow[col - kNodeDim];
        a.y = erow[col - kNodeDim + 1];
      }
#pragma unroll
      for (int nt = 0; nt < 4; ++nt) {
        const float* wp = &sW1[l][nt * 16 + lrow][col];
        v2f b;
        b.x = wp[0];
        b.y = wp[1];
        hacc[nt] = wmma_f32(a, b, hacc[nt]);
      }
    }

    // ReLU + C/D-layout -> A-layout conversion through this wave's LDS tile.
#pragma unroll
    for (int nt = 0; nt < 4; ++nt)
#pragma unroll
      for (int r = 0; r < 8; ++r) {
        float v = hacc[nt][r];
        sH[wave][r + khalf * 8][nt * 16 + lrow] = v > 0.0f ? v : 0.0f;
      }

    // GEMM2: [16 x 64] @ [64 x 64], accumulating into macc.
#pragma unroll
    for (int ks = 0; ks < kHidden / 4; ++ks) {
      const int col = ks * 4 + koff;
      v2f a;
      a.x = sH[wave][lrow][col];
      a.y = sH[wave][lrow][col + 1];
#pragma unroll
      for (int nt = 0; nt < 4; ++nt) {
        const float* wp = &sW2[l][nt * 16 + lrow][col];
        v2f b;
        b.x = wp[0];
        b.y = wp[1];
        macc[nt] = wmma_f32(a, b, macc[nt]);
      }
    }
  }

  // Scatter-add the 16x64 message tile into agg (L2-resident, 12.8 MB).
#pragma unroll
  for (int r = 0; r < 8; ++r) {
    const int d = sDst[wave][r + khalf * 8];
    float* arow = agg + (long)d * kHidden;
#pragma unroll
    for (int nt = 0; nt < 4; ++nt)
      atomicAdd(arow + nt * 16 + lrow, macc[nt][r]);
  }
}

// One wave handles 16 nodes: out = relu([node|aggN]@U1 + c1) @ U2 + c2.
__global__ __launch_bounds__(kThreads) void gmp_node_kernel(
    const float* __restrict__ node_f, const float* __restrict__ aggN,
    const float* __restrict__ upd_w1, const float* __restrict__ upd_b1,
    const float* __restrict__ upd_w2, const float* __restrict__ upd_b2,
    float* __restrict__ out) {
  __shared__ float sW1[kHidden][kU1Stride];  // transposed: [n][k]
  __shared__ float sW2[kHidden][kW2Stride];  // transposed: [n][k]
  __shared__ float sB1[kHidden];
  __shared__ float sB2[kHidden];
  __shared__ float sH[kWaves][16][kHStride];

  const int tid = threadIdx.x;
  for (int i = tid; i < kInUpd * kHidden; i += kThreads) {
    int k = i / kHidden;
    int n = i - k * kHidden;
    sW1[n][k] = upd_w1[i];
  }
  for (int i = tid; i < kHidden * kHidden; i += kThreads) {
    int k = i / kHidden;
    int n = i - k * kHidden;
    sW2[n][k] = upd_w2[i];
  }
  if (tid < kHidden) {
    sB1[tid] = upd_b1[tid];
    sB2[tid] = upd_b2[tid];
  }
  __syncthreads();

  const int wave  = tid >> 5;
  const int lane  = tid & 31;
  const int lrow  = lane & 15;
  const int khalf = lane >> 4;
  const int koff  = khalf * 2;

  const long tile = (long)blockIdx.x * kWaves + wave;
  if (tile * 16 >= kNodes) return;  // wave-uniform: EXEC stays all-ones
  const long nbase = tile * 16;

  const float* nrow = node_f + (nbase + lrow) * kNodeDim;
  const float* arow = aggN + (nbase + lrow) * kHidden;

  v8f hacc[4];
#pragma unroll
  for (int nt = 0; nt < 4; ++nt) {
    float b = sB1[nt * 16 + lrow];
#pragma unroll
    for (int j = 0; j < 8; ++j) hacc[nt][j] = b;
  }

  // GEMM1: [16 x 128] @ [128 x 64]
#pragma unroll
  for (int ks = 0; ks < kInUpd / 4; ++ks) {
    const int col = ks * 4 + koff;
    v2f a;
    if (col < kNodeDim) {
      a.x = nrow[col];
      a.y = nrow[col + 1];
    } else {
      a.x = arow[col - kNodeDim];
      a.y = arow[col - kNodeDim + 1];
    }
#pragma unroll
    for (int nt = 0; nt < 4; ++nt) {
      const float* wp = &sW1[nt * 16 + lrow][col];
      v2f b;
      b.x = wp[0];
      b.y = wp[1];
      hacc[nt] = wmma_f32(a, b, hacc[nt]);
    }
  }

#pragma unroll
  for (int nt = 0; nt < 4; ++nt)
#pragma unroll
    for (int r = 0; r < 8; ++r) {
      float v = hacc[nt][r];
      sH[wave][r + khalf * 8][nt * 16 + lrow] = v > 0.0f ? v : 0.0f;
    }

  v8f macc[4];
#pragma unroll
  for (int nt = 0; nt < 4; ++nt) {
    float b = sB2[nt * 16 + lrow];
#pragma unroll
    for (int j = 0; j < 8; ++j) macc[nt][j] = b;
  }

  // GEMM2: [16 x 64] @ [64 x 64]
#pragma unroll
  for (int ks = 0; ks < kHidden / 4; ++ks) {
    const int col = ks * 4 + koff;
    v2f a;
    a.x = sH[wave][lrow][col];
    a.y = sH[wave][lrow][col + 1];
#pragma unroll
    for (int nt = 0; nt < 4; ++nt) {
      const float* wp = &sW2[nt * 16 + lrow][col];
      v2f b;
      b.x = wp[0];
      b.y = wp[1];
      macc[nt] = wmma_f32(a, b, macc[nt]);
    }
  }

  // Store the 16x64 output tile.
#pragma unroll
  for (int r = 0; r < 8; ++r) {
    const long nidx = nbase + r + khalf * 8;
    float* orow = out + nidx * kNodeDim;
#pragma unroll
    for (int nt = 0; nt < 4; ++nt) orow[nt * 16 + lrow] = macc[nt][r];
  }
}

extern "C" void kernel_launch(void* const* d_in, const int* in_sizes, int n_in,
                              void* d_out, int out_size, void* d_ws,
                              size_t ws_size, hipStream_t stream) {
  (void)in_sizes;
  (void)n_in;
  (void)out_size;
  (void)ws_size;
  const float* node_f = (const float*)d_in[0];
  const float* edge_f = (const float*)d_in[1];
  const float* msg_w1 = (const float*)d_in[2];
  const float* msg_b1 = (const float*)d_in[3];
  const float* msg_w2 = (const float*)d_in[4];
  const float* msg_b2 = (const float*)d_in[5];
  const float* upd_w1 = (const float*)d_in[6];
  const float* upd_b1 = (const float*)d_in[7];
  const float* upd_w2 = (const float*)d_in[8];
  const float* upd_b2 = (const float*)d_in[9];
  const int* src = (const int*)d_in[10];
  const int* dst = (const int*)d_in[11];
  float* out = (float*)d_out;

  float* agg = (float*)d_ws;                      // [kNodes][kHidden]
  float* deg = agg + (size_t)kNodes * kHidden;    // [kNodes]

  const int ztotal = kNodes * kHidden + kNodes;
  gmp_zero_kernel<<<(ztotal + 255) / 256, 256, 0, stream>>>(agg, ztotal);

  gmp_edge_kernel<<<kEdges / (kWaves * 16), kThreads, 0, stream>>>(
      node_f, edge_f, msg_w1, msg_b1, msg_w2, msg_b2, src, dst, agg, deg);

  gmp_norm_kernel<<<(kNodes * kHidden + 255) / 256, 256, 0, stream>>>(agg, deg);

  gmp_node_kernel<<<(kNodes / 16 + kWaves - 1) / kWaves, kThreads, 0, stream>>>(
      node_f, agg, upd_w1, upd_b1, upd_w2, upd_b2, out);
}